// SAGNet_64089501991545
// MI455X (gfx1250) — compile-verified
//
#include <hip/hip_runtime.h>
#include <math.h>

#define B_GRAPHS 50
#define NPG0     2000
#define EPG      12000
#define E_TOTAL  (B_GRAPHS * EPG)   // 600000
#define F_IN     256
#define NH       128
#define SORT_N   2048

typedef __attribute__((ext_vector_type(2))) float v2f;
typedef __attribute__((ext_vector_type(8))) float v8f;

// ---------------------------------------------------------------------------
// Dense GEMM  C[M,128] = A[M,K] @ B[K,128]  using V_WMMA_F32_16X16X4_F32.
// One wave owns a full 16x128 output strip. Per K-step: one A fragment
// (reused 8x) and 8 independent B fragments loaded up-front into distinct
// registers so the loads clause/pipeline instead of serializing each WMMA.
// 8 waves (8 strips) per 256-thread block.
// ---------------------------------------------------------------------------
__global__ void gemm_wmma(const float* __restrict__ A, const float* __restrict__ Bm,
                          float* __restrict__ C, int M, int K) {
    const int NN = NH;                       // 128 (8 column tiles)
    int lane = threadIdx.x & 31;
    int wave = threadIdx.x >> 5;
    int strip = blockIdx.x * 8 + wave;       // 16-row strip index
    if (strip >= (M >> 4)) return;           // wave-uniform exit (EXEC all-1s for WMMA)
    int row = strip << 4;
    int l15 = lane & 15;
    int kHalf = (lane >> 4) << 1;            // 0 for lanes 0-15, 2 for lanes 16-31
    int aRow = row + l15;

    v8f acc[8];
#pragma unroll
    for (int t = 0; t < 8; ++t) acc[t] = (v8f){};

    for (int k = 0; k < K; k += 4) {
        // A fragment (16x4): lane m=l15, VGPR0 = K=k+kHalf, VGPR1 = K=k+kHalf+1
        const float* ap = A + (size_t)aRow * K + (k + kHalf);
        v2f a; a.x = ap[0]; a.y = ap[1];
        // B fragments (4x16 each): lanes 0-15 rows k+0/k+1, lanes 16-31 rows k+2/k+3
        // Load all 8 into distinct registers first -> one clause, staged waits.
        const float* bbase = Bm + (size_t)(k + kHalf) * NN + l15;
        v2f bfrag[8];
#pragma unroll
        for (int t = 0; t < 8; ++t) {
            bfrag[t].x = bbase[t * 16];
            bfrag[t].y = bbase[t * 16 + NN];
        }
#pragma unroll
        for (int t = 0; t < 8; ++t) {
            acc[t] = __builtin_amdgcn_wmma_f32_16x16x4_f32(
                false, a, false, bfrag[t], (short)0, acc[t], false, false);
        }
    }
    // D layout: VGPR j -> M = row + j + 8*(lane>=16), N = t*16 + (lane&15)
    int mBase = row + ((lane >> 4) << 3);
#pragma unroll
    for (int t = 0; t < 8; ++t) {
        float* cp = C + (size_t)mBase * NN + t * 16 + l15;
#pragma unroll
        for (int j = 0; j < 8; ++j) cp[(size_t)j * NN] = acc[t][j];
    }
}

// ---------------------------------------------------------------------------
// Degree / norm helpers
// ---------------------------------------------------------------------------
__global__ void deg_accum(const int* __restrict__ dst, const float* __restrict__ w,
                          float* __restrict__ deg, int E) {
    int e = blockIdx.x * blockDim.x + threadIdx.x;
    if (e >= E) return;
    float wv = w ? w[e] : 1.0f;
    if (wv != 0.0f) atomicAdd(&deg[dst[e]], wv);
}

__global__ void finalize_deg(float* __restrict__ deg, float* __restrict__ dis, int N) {
    int i = blockIdx.x * blockDim.x + threadIdx.x;
    if (i >= N) return;
    float d = deg[i] + 1.0f;    // self loop
    deg[i] = d;
    dis[i] = rsqrtf(d);
}

__global__ void norm_kernel(const int* __restrict__ src, const int* __restrict__ dst,
                            const float* __restrict__ w, const float* __restrict__ dis,
                            float* __restrict__ nrm, int E) {
    int e = blockIdx.x * blockDim.x + threadIdx.x;
    if (e >= E) return;
    float wv = w ? w[e] : 1.0f;
    nrm[e] = wv * dis[src[e]] * dis[dst[e]];
}

// out[i,f] = H[i,f]/deg[i] + b[f]   (self-loop contribution + bias), then atomics add
__global__ void selfloop_init(const float* __restrict__ H, const float* __restrict__ deg,
                              const float* __restrict__ b, float* __restrict__ out, int N) {
    long long i = (long long)blockIdx.x * blockDim.x + threadIdx.x;
    if (i >= (long long)N * NH) return;
    int node = (int)(i >> 7);
    int f = (int)(i & (NH - 1));
    out[i] = H[i] / deg[node] + b[f];
}

__global__ void edge_agg(const float* __restrict__ H, const int* __restrict__ src,
                         const int* __restrict__ dst, const float* __restrict__ nrm,
                         float* __restrict__ out, int E) {
    long long idx = (long long)blockIdx.x * blockDim.x + threadIdx.x;
    if (idx >= (long long)E * 32) return;
    int e = (int)(idx >> 5);
    int f = (int)(idx & 31) * 4;
    float nv = nrm[e];
    if (nv == 0.0f) return;                 // dropped edge
    const float* hp = H + (size_t)src[e] * NH + f;
    float* op = out + (size_t)dst[e] * NH + f;
    atomicAdd(op + 0, hp[0] * nv);
    atomicAdd(op + 1, hp[1] * nv);
    atomicAdd(op + 2, hp[2] * nv);
    atomicAdd(op + 3, hp[3] * nv);
}

__global__ void relu_k(float* __restrict__ x, long long n) {
    long long i = (long long)blockIdx.x * blockDim.x + threadIdx.x;
    if (i < n) x[i] = fmaxf(x[i], 0.0f);
}

// ---------------------------------------------------------------------------
// Score path: hsc = X @ Ws (128 -> 1), then GCN-normalized aggregation
// ---------------------------------------------------------------------------
__global__ void gemv128(const float* __restrict__ X, const float* __restrict__ Ws,
                        float* __restrict__ out, int N) {
    int gtid = blockIdx.x * blockDim.x + threadIdx.x;
    int row = gtid >> 5;
    int lane = threadIdx.x & 31;
    if (row >= N) return;
    const float* xr = X + (size_t)row * NH;
    float acc = 0.0f;
#pragma unroll
    for (int f = lane; f < NH; f += 32) acc += xr[f] * Ws[f];
    for (int off = 16; off > 0; off >>= 1) acc += __shfl_down(acc, off, 32);
    if (lane == 0) out[row] = acc;
}

__global__ void score_init(const float* __restrict__ hsc, const float* __restrict__ deg,
                           const float* __restrict__ bs, float* __restrict__ score, int N) {
    int i = blockIdx.x * blockDim.x + threadIdx.x;
    if (i < N) score[i] = hsc[i] / deg[i] + bs[0];
}

__global__ void score_edge(const float* __restrict__ hsc, const int* __restrict__ src,
                           const int* __restrict__ dst, const float* __restrict__ nrm,
                           float* __restrict__ score, int E) {
    int e = blockIdx.x * blockDim.x + threadIdx.x;
    if (e >= E) return;
    float nv = nrm[e];
    if (nv != 0.0f) atomicAdd(&score[dst[e]], hsc[src[e]] * nv);
}

// ---------------------------------------------------------------------------
// Per-graph top-k: bitonic sort of 2048 (score, idx) pairs in LDS, descending,
// tie-break on lower index (matches lax.top_k stability).
// ---------------------------------------------------------------------------
__global__ void topk_kernel(const float* __restrict__ score, int n, int k,
                            int* __restrict__ perm) {
    __shared__ float s[SORT_N];
    __shared__ int si[SORT_N];
    int g = blockIdx.x;
    for (int i = threadIdx.x; i < SORT_N; i += blockDim.x) {
        if (i < n) { s[i] = score[g * n + i]; si[i] = i; }
        else { s[i] = -INFINITY; si[i] = SORT_N + i; }
    }
    __syncthreads();
    for (int size = 2; size <= SORT_N; size <<= 1) {
        for (int stride = size >> 1; stride > 0; stride >>= 1) {
            int t = threadIdx.x;                              // 0..1023
            int pos = 2 * t - (t & (stride - 1));
            bool descend = ((pos & size) == 0);
            float a = s[pos], b = s[pos + stride];
            int ia = si[pos], ib = si[pos + stride];
            bool aFirst = (a > b) || (a == b && ia < ib);
            if (aFirst != descend) {
                s[pos] = b; s[pos + stride] = a;
                si[pos] = ib; si[pos + stride] = ia;
            }
            __syncthreads();
        }
    }
    for (int i = threadIdx.x; i < k; i += blockDim.x)
        perm[g * k + i] = g * n + si[i];
}

__global__ void gather_k(const float* __restrict__ Xmid, const float* __restrict__ score,
                         const int* __restrict__ perm, float* __restrict__ Xout, int M) {
    long long idx = (long long)blockIdx.x * blockDim.x + threadIdx.x;
    if (idx >= (long long)M * 32) return;
    int i = (int)(idx >> 5);
    int f = (int)(idx & 31) * 4;
    int p = perm[i];
    float t = tanhf(score[p]);
    const float* sp = Xmid + (size_t)p * NH + f;
    float* op = Xout + (size_t)i * NH + f;
    op[0] = sp[0] * t; op[1] = sp[1] * t; op[2] = sp[2] * t; op[3] = sp[3] * t;
}

__global__ void set_mapping(const int* __restrict__ perm, int* __restrict__ mapping, int M) {
    int i = blockIdx.x * blockDim.x + threadIdx.x;
    if (i < M) mapping[perm[i]] = i;
}

__global__ void filter_edges(const int* __restrict__ src, const int* __restrict__ dst,
                             const float* __restrict__ w, const int* __restrict__ mapping,
                             int* __restrict__ srcO, int* __restrict__ dstO,
                             float* __restrict__ wO, int E) {
    int e = blockIdx.x * blockDim.x + threadIdx.x;
    if (e >= E) return;
    int ns = mapping[src[e]];
    int nd = mapping[dst[e]];
    float wv = w ? w[e] : 1.0f;
    bool valid = (ns >= 0) && (nd >= 0) && (wv > 0.0f);
    srcO[e] = valid ? ns : 0;
    dstO[e] = valid ? nd : 0;
    wO[e] = valid ? wv : 0.0f;
}

// readout: xsum[g, f] += max over k rows ; xsum[g, 128+f] += mean
__global__ void readout_k(const float* __restrict__ X, int k, float* __restrict__ xsum) {
    int g = blockIdx.x, f = threadIdx.x;  // 128 threads
    float mx = -INFINITY, sm = 0.0f;
    const float* base = X + (size_t)g * k * NH + f;
    for (int r = 0; r < k; ++r) {
        float v = base[(size_t)r * NH];
        mx = fmaxf(mx, v);
        sm += v;
    }
    xsum[g * 2 * NH + f] += mx;
    xsum[g * 2 * NH + NH + f] += sm / (float)k;
}

__global__ void final_linear(const float* __restrict__ xsum, const float* __restrict__ Wl,
                             const float* __restrict__ bl, float* __restrict__ out) {
    int g = blockIdx.x, j = threadIdx.x;  // 128 threads
    float acc = bl[j];
    const float* xr = xsum + g * 2 * NH;
#pragma unroll 4
    for (int f = 0; f < 2 * NH; ++f) acc += xr[f] * Wl[f * NH + j];
    out[g * NH + j] = fmaxf(acc, 0.0f);
}

// ---------------------------------------------------------------------------
// Host-side round driver (stream-ordered; graph-capture safe)
// ---------------------------------------------------------------------------
static void run_round(hipStream_t stream,
                      const float* x_in, int K,
                      const float* W, const float* b, const float* Ws, const float* bs,
                      const int* src, const int* dst, const float* w,   // w==null -> ones
                      int n, int k,
                      float* H, float* Xmid, float* Xout,
                      float* deg, float* dis, float* hsc, float* sco, float* nrm,
                      int* mapping, int* perm,
                      int* srcO, int* dstO, float* wO,                  // null -> skip filter
                      float* xsum) {
    const int E = E_TOTAL;
    int N = B_GRAPHS * n;

    // degrees + norms
    hipMemsetAsync(deg, 0, (size_t)N * sizeof(float), stream);
    deg_accum<<<(E + 255) / 256, 256, 0, stream>>>(dst, w, deg, E);
    finalize_deg<<<(N + 255) / 256, 256, 0, stream>>>(deg, dis, N);
    norm_kernel<<<(E + 255) / 256, 256, 0, stream>>>(src, dst, w, dis, nrm, E);

    // H = x_in @ W  (WMMA, one 16x128 strip per wave)
    int strips = N / 16;
    gemm_wmma<<<(strips + 7) / 8, 256, 0, stream>>>(x_in, W, H, N, K);

    // Xmid = segsum(H[src]*norm) + H/deg + b ; relu
    long long nf = (long long)N * NH;
    selfloop_init<<<(unsigned)((nf + 255) / 256), 256, 0, stream>>>(H, deg, b, Xmid, N);
    edge_agg<<<(unsigned)(((long long)E * 32 + 255) / 256), 256, 0, stream>>>(H, src, dst, nrm, Xmid, E);
    relu_k<<<(unsigned)((nf + 255) / 256), 256, 0, stream>>>(Xmid, nf);

    // scoring GCN (128 -> 1)
    gemv128<<<(N * 32 + 255) / 256, 256, 0, stream>>>(Xmid, Ws, hsc, N);
    score_init<<<(N + 255) / 256, 256, 0, stream>>>(hsc, deg, bs, sco, N);
    score_edge<<<(E + 255) / 256, 256, 0, stream>>>(hsc, src, dst, nrm, sco, E);

    // top-k + gather
    topk_kernel<<<B_GRAPHS, 1024, 0, stream>>>(sco, n, k, perm);
    int M = B_GRAPHS * k;
    gather_k<<<(unsigned)(((long long)M * 32 + 255) / 256), 256, 0, stream>>>(Xmid, sco, perm, Xout, M);

    // readout accumulate
    readout_k<<<B_GRAPHS, NH, 0, stream>>>(Xout, k, xsum);

    // edge filtering for next round
    if (srcO) {
        hipMemsetAsync(mapping, 0xFF, (size_t)N * sizeof(int), stream);  // -1
        set_mapping<<<(M + 255) / 256, 256, 0, stream>>>(perm, mapping, M);
        filter_edges<<<(E + 255) / 256, 256, 0, stream>>>(src, dst, w, mapping, srcO, dstO, wO, E);
    }
}

extern "C" void kernel_launch(void* const* d_in, const int* in_sizes, int n_in,
                              void* d_out, int out_size, void* d_ws, size_t ws_size,
                              hipStream_t stream) {
    const float* x0  = (const float*)d_in[0];
    const float* W1  = (const float*)d_in[1];
    const float* b1  = (const float*)d_in[2];
    const float* Ws1 = (const float*)d_in[3];
    const float* bs1 = (const float*)d_in[4];
    const float* W2  = (const float*)d_in[5];
    const float* b2  = (const float*)d_in[6];
    const float* Ws2 = (const float*)d_in[7];
    const float* bs2 = (const float*)d_in[8];
    const float* W3  = (const float*)d_in[9];
    const float* b3  = (const float*)d_in[10];
    const float* Ws3 = (const float*)d_in[11];
    const float* bs3 = (const float*)d_in[12];
    const float* Wl  = (const float*)d_in[13];
    const float* bl  = (const float*)d_in[14];
    const int* eidx  = (const int*)d_in[15];
    const int* src0 = eidx;
    const int* dst0 = eidx + E_TOTAL;
    (void)in_sizes; (void)n_in; (void)out_size; (void)ws_size;

    // workspace carve-out
    char* p = (char*)d_ws;
    auto carve = [&](size_t bytes) { char* r = p; p += (bytes + 255) & ~(size_t)255; return (void*)r; };
    const size_t NMAX = (size_t)B_GRAPHS * NPG0;                 // 100000
    float* H    = (float*)carve(NMAX * NH * sizeof(float));      // 51.2 MB
    float* X1   = (float*)carve(NMAX * NH * sizeof(float));      // 51.2 MB (Xmid)
    float* X2   = (float*)carve((size_t)80000 * NH * sizeof(float)); // 41 MB (Xout / x_in)
    float* deg  = (float*)carve(NMAX * sizeof(float));
    float* dis  = (float*)carve(NMAX * sizeof(float));
    float* hsc  = (float*)carve(NMAX * sizeof(float));
    float* sco  = (float*)carve(NMAX * sizeof(float));
    float* nrm  = (float*)carve((size_t)E_TOTAL * sizeof(float));
    int*   srcA = (int*)carve((size_t)E_TOTAL * sizeof(int));
    int*   dstA = (int*)carve((size_t)E_TOTAL * sizeof(int));
    float* wA   = (float*)carve((size_t)E_TOTAL * sizeof(float));
    int*   srcB = (int*)carve((size_t)E_TOTAL * sizeof(int));
    int*   dstB = (int*)carve((size_t)E_TOTAL * sizeof(int));
    float* wB   = (float*)carve((size_t)E_TOTAL * sizeof(float));
    int*   mapping = (int*)carve(NMAX * sizeof(int));
    int*   perm    = (int*)carve((size_t)80000 * sizeof(int));
    float* xsum    = (float*)carve((size_t)B_GRAPHS * 2 * NH * sizeof(float));

    hipMemsetAsync(xsum, 0, (size_t)B_GRAPHS * 2 * NH * sizeof(float), stream);

    // round 1: n=2000, k=1600, F_in=256, w = ones
    run_round(stream, x0, F_IN, W1, b1, Ws1, bs1, src0, dst0, nullptr,
              NPG0, 1600, H, X1, X2, deg, dis, hsc, sco, nrm,
              mapping, perm, srcA, dstA, wA, xsum);
    // round 2: n=1600, k=1280  (x_in = X2; safe: GEMM reads X2 before gather rewrites it)
    run_round(stream, X2, NH, W2, b2, Ws2, bs2, srcA, dstA, wA,
              1600, 1280, H, X1, X2, deg, dis, hsc, sco, nrm,
              mapping, perm, srcB, dstB, wB, xsum);
    // round 3: n=1280, k=1024 ; filtered edges unused afterwards -> skip
    run_round(stream, X2, NH, W3, b3, Ws3, bs3, srcB, dstB, wB,
              1280, 1024, H, X1, X2, deg, dis, hsc, sco, nrm,
              mapping, perm, nullptr, nullptr, nullptr, xsum);

    // out = relu((x1+x2+x3) @ Wl + bl)   [50 x 128]
    final_linear<<<B_GRAPHS, NH, 0, stream>>>(xsum, Wl, bl, (float*)d_out);
}